// NeuralProgrammer_74139725463588
// MI455X (gfx1250) — compile-verified
//
#include <hip/hip_runtime.h>
#include <math.h>

#define H      256
#define QLEN   64
#define NNUM   8
#define NOPS   9
#define NCOLS  16
#define TSTEPS 4
#define NROWS  1000000

#define OP_SUM 0
#define OP_COUNT 1
#define OP_DIFF 2
#define OP_GREATER 3
#define OP_LESSER 4
#define OP_AND 5
#define OP_OR 6
#define OP_ASSIGN 7
#define OP_RESET 8

// workspace layout (float offsets)
#define WS_RED  0                // 8: S[0..3], W[0..3]
#define WS_PIV  8                // 2: l_pivot, g_pivot
#define WS_AOP  16               // 36: a_op[4][9]
#define WS_ACOL 64               // 64: a_col[4][16]
#define WS_XW   128              // 16384: xs @ Wx^T
#define WS_HS   (128 + 16384)    // 16384: question RNN hidden states

typedef __attribute__((ext_vector_type(2))) float v2f;
typedef __attribute__((ext_vector_type(8))) float v8f;

__global__ void np_init(float* ws) {
    if (threadIdx.x < 8) ws[WS_RED + threadIdx.x] = 0.0f;
}

// XW[m][n] = sum_k emb[iq[m]][k] * Wx[n][k], via V_WMMA_F32_16X16X4_F32.
// One wave (32 lanes) per 16x16 output tile; grid = (64/16)*(256/16) = 64 tiles.
__global__ void np_gemm_xw(const int* __restrict__ iq, const float* __restrict__ emb,
                           const float* __restrict__ Wx, float* __restrict__ ws) {
    int tile = blockIdx.x;
    int mt = tile >> 4;            // 0..3   (M tiles over Q=64)
    int nt = tile & 15;            // 0..15  (N tiles over H=256)
    int lane = threadIdx.x;        // 0..31
    int half = lane >> 4;          // 0 or 1 (selects K=0,1 vs K=2,3 in the frag)
    int l15  = lane & 15;

    int m = mt * 16 + l15;                 // A row for this lane
    int n = nt * 16 + l15;                 // B col for this lane
    const float* arow = emb + (long)iq[m] * H;
    const float* brow = Wx  + (long)n * H; // B[k][n] = Wx[n][k]

    v8f acc = {0.f, 0.f, 0.f, 0.f, 0.f, 0.f, 0.f, 0.f};
    for (int k0 = 0; k0 < H; k0 += 4) {
        int kb = k0 + half * 2;
        v2f a; a.x = arow[kb]; a.y = arow[kb + 1];  // A 16x4: V0=K0/K2, V1=K1/K3
        v2f b; b.x = brow[kb]; b.y = brow[kb + 1];  // B 4x16: same K split across lane halves
        acc = __builtin_amdgcn_wmma_f32_16x16x4_f32(
            false, a, false, b, (short)0, acc, false, false);
    }
    // D layout: VGPR i -> M = i + 8*half, N = nt*16 + l15
    float* XW = ws + WS_XW;
    int mrow = mt * 16 + half * 8;
#pragma unroll
    for (int i = 0; i < 8; ++i)
        XW[(mrow + i) * H + nt * 16 + l15] = acc[i];
}

// Sequential tanh RNN over the question: h_t = tanh(XW[t] + h_{t-1} @ Wh^T)
__global__ void np_rnn(const float* __restrict__ Wh, float* __restrict__ ws) {
    __shared__ float h[H], h2[H];
    int j = threadIdx.x;
    const float* XW = ws + WS_XW;
    float* hs = ws + WS_HS;
    h[j] = 0.0f;
    __syncthreads();
    const float* wrow = Wh + j * H;
    for (int s = 0; s < QLEN; ++s) {
        float acc = XW[s * H + j];
        for (int k = 0; k < H; ++k) acc += h[k] * wrow[k];
        float v = tanhf(acc);
        hs[s * H + j] = v;
        h2[j] = v;
        __syncthreads();
        h[j] = h2[j];
        __syncthreads();
    }
}

// Pivots + the full T=4 selector/history unroll (table-independent).
__global__ void np_select(const int* __restrict__ nums_i, const int* __restrict__ lwi,
                          const float* __restrict__ W_op, const float* __restrict__ op_emb,
                          const float* __restrict__ W_col, const float* __restrict__ col_emb,
                          const float* __restrict__ W_hist, const float* __restrict__ U,
                          float* __restrict__ ws) {
    __shared__ float q[H], hh[H], tmp[H], p1[H], p2[H];
    __shared__ float lop[NOPS], aop_s[NOPS], lcol[NCOLS], acol_s[NCOLS];
    __shared__ float zdot[2][NNUM];
    int j = threadIdx.x;
    const float* hs = ws + WS_HS;
    q[j]  = hs[(QLEN - 1) * H + j];
    hh[j] = 0.0f;
    if (j < 16) {                      // 16 dot products z_i . U_u
        int u = j >> 3, i = j & 7;
        const float* z  = hs + (long)lwi[i] * H;
        const float* uv = U + u * H;
        float d = 0.0f;
        for (int k = 0; k < H; ++k) d += z[k] * uv[k];
        zdot[u][i] = d;
    }
    __syncthreads();
    if (j < 2) {                       // softmax over 8, dot with question numbers
        float mx = zdot[j][0];
        for (int i = 1; i < NNUM; ++i) mx = fmaxf(mx, zdot[j][i]);
        float den = 0.0f, num = 0.0f;
        for (int i = 0; i < NNUM; ++i) {
            float e = __expf(zdot[j][i] - mx);
            den += e; num += e * (float)nums_i[i];
        }
        ws[WS_PIV + j] = num / den;    // [0]=l_pivot (U0), [1]=g_pivot (U1)
    }
    for (int t = 0; t < TSTEPS; ++t) {
        // op branch: softmax(op_emb @ tanh(W_op @ [q; hh]))
        const float* w = W_op + (long)j * (2 * H);
        float acc = 0.0f;
        for (int k = 0; k < H; ++k) acc += w[k] * q[k];
        for (int k = 0; k < H; ++k) acc += w[H + k] * hh[k];
        tmp[j] = tanhf(acc);
        __syncthreads();
        if (j < NOPS) {
            const float* oe = op_emb + j * H;
            float d = 0.0f;
            for (int k = 0; k < H; ++k) d += oe[k] * tmp[k];
            lop[j] = d;
        }
        __syncthreads();
        if (j == 0) {
            float mx = lop[0];
            for (int o = 1; o < NOPS; ++o) mx = fmaxf(mx, lop[o]);
            float den = 0.0f, e[NOPS];
            for (int o = 0; o < NOPS; ++o) { e[o] = __expf(lop[o] - mx); den += e[o]; }
            for (int o = 0; o < NOPS; ++o) {
                aop_s[o] = e[o] / den;
                ws[WS_AOP + t * NOPS + o] = aop_s[o];
            }
        }
        __syncthreads();
        // column branch
        const float* wc = W_col + (long)j * (2 * H);
        acc = 0.0f;
        for (int k = 0; k < H; ++k) acc += wc[k] * q[k];
        for (int k = 0; k < H; ++k) acc += wc[H + k] * hh[k];
        tmp[j] = tanhf(acc);
        __syncthreads();
        if (j < NCOLS) {
            const float* ce = col_emb + j * H;
            float d = 0.0f;
            for (int k = 0; k < H; ++k) d += ce[k] * tmp[k];
            lcol[j] = d;
        }
        __syncthreads();
        if (j == 0) {
            float mx = lcol[0];
            for (int c = 1; c < NCOLS; ++c) mx = fmaxf(mx, lcol[c]);
            float den = 0.0f, e[NCOLS];
            for (int c = 0; c < NCOLS; ++c) { e[c] = __expf(lcol[c] - mx); den += e[c]; }
            for (int c = 0; c < NCOLS; ++c) {
                acol_s[c] = e[c] / den;
                ws[WS_ACOL + t * NCOLS + c] = acol_s[c];
            }
        }
        __syncthreads();
        // history RNN: hh = tanh(W_hist @ [aop@op_emb; acol@col_emb; hh])
        float a1 = 0.0f, a2 = 0.0f;
        for (int o = 0; o < NOPS; ++o)  a1 += aop_s[o]  * op_emb[o * H + j];
        for (int c = 0; c < NCOLS; ++c) a2 += acol_s[c] * col_emb[c * H + j];
        p1[j] = a1; p2[j] = a2;
        __syncthreads();
        const float* wh = W_hist + (long)j * (3 * H);
        acc = 0.0f;
        for (int k = 0; k < H; ++k) acc += wh[k] * p1[k];
        for (int k = 0; k < H; ++k) acc += wh[H + k] * p2[k];
        for (int k = 0; k < H; ++k) acc += wh[2 * H + k] * hh[k];
        tmp[j] = tanhf(acc);
        __syncthreads();
        hh[j] = tmp[j];
        __syncthreads();
    }
}

// The single streaming pass over the table (memory-bound: 64 MB in, 64 MB out).
#define BLK 256
__global__ __launch_bounds__(BLK)
void np_table(const float* __restrict__ table, float* __restrict__ ws,
              float* __restrict__ out) {
    __shared__ float acol_l[TSTEPS * NCOLS];
    __shared__ float aop_l[TSTEPS * NOPS];
    __shared__ float piv[2];
    __shared__ float stage[BLK * NCOLS];   // 16KB lookup staging for coalesced stores
    __shared__ float red_s[8][BLK];        // block reduction
    int tid = threadIdx.x;
    if (tid < TSTEPS * NCOLS) acol_l[tid] = ws[WS_ACOL + tid];
    if (tid < TSTEPS * NOPS)  aop_l[tid]  = ws[WS_AOP + tid];
    if (tid < 2)              piv[tid]    = ws[WS_PIV + tid];
    __syncthreads();

    float lp = piv[0], gp = piv[1];
    long r = (long)blockIdx.x * BLK + tid;
    bool act = (r < (long)NROWS);
    float contribS[TSTEPS] = {0.f, 0.f, 0.f, 0.f};
    float contribW[TSTEPS] = {0.f, 0.f, 0.f, 0.f};
    float rs_fin = 0.0f;

    if (act) {
        const float4* tp = (const float4*)(table + r * NCOLS);
        float4 v0 = tp[0], v1 = tp[1], v2 = tp[2], v3 = tp[3];
        float tv[NCOLS] = {v0.x, v0.y, v0.z, v0.w, v1.x, v1.y, v1.z, v1.w,
                           v2.x, v2.y, v2.z, v2.w, v3.x, v3.y, v3.z, v3.w};
        float rowsum = 0.0f;
#pragma unroll
        for (int c = 0; c < NCOLS; ++c) rowsum += tv[c];
        float sl[NCOLS], sg[NCOLS];
#pragma unroll
        for (int c = 0; c < NCOLS; ++c) {
            sl[c] = 1.0f / (1.0f + __expf(tv[c] - lp));  // sigmoid(l_pivot - t)
            sg[c] = 1.0f / (1.0f + __expf(gp - tv[c]));  // sigmoid(t - g_pivot)
        }
        float rs1 = 1.0f, rs2 = 1.0f;
#pragma unroll
        for (int t = 0; t < TSTEPS; ++t) {
            contribS[t] = rs1;
            contribW[t] = rs1 * rowsum;
            float dg = 0.0f, dl = 0.0f;
#pragma unroll
            for (int c = 0; c < NCOLS; ++c) {
                float ac = acol_l[t * NCOLS + c];
                dg += sg[c] * ac;
                dl += sl[c] * ac;
            }
            const float* ao = aop_l + t * NOPS;
            float keep = ao[OP_SUM] + ao[OP_COUNT] + ao[OP_DIFF] + ao[OP_ASSIGN];
            float rs = ao[OP_GREATER] * dg + ao[OP_LESSER] * dl
                     + ao[OP_AND] * fminf(rs1, rs2) + ao[OP_OR] * fmaxf(rs1, rs2)
                     + ao[OP_RESET] + keep * rs1;
            rs2 = rs1; rs1 = rs;
        }
        rs_fin = rs1;
    }

    // stage lookup row: aop[3][ASSIGN] * rs_4[r] * acol[3][c]
    float lv = aop_l[3 * NOPS + OP_ASSIGN] * rs_fin;
#pragma unroll
    for (int c = 0; c < NCOLS; ++c)
        stage[tid * NCOLS + c] = lv * acol_l[3 * NCOLS + c];

    // block-reduce the 8 scalars
#pragma unroll
    for (int v = 0; v < TSTEPS; ++v) {
        red_s[v][tid]     = contribS[v];
        red_s[4 + v][tid] = contribW[v];
    }
    __syncthreads();
    for (int off = BLK / 2; off > 0; off >>= 1) {
        if (tid < off)
#pragma unroll
            for (int v = 0; v < 8; ++v) red_s[v][tid] += red_s[v][tid + off];
        __syncthreads();
    }
    if (tid < 8) atomicAdd(ws + WS_RED + tid, red_s[tid][0]);

    // coalesced dword stores of the staged lookup (d_out+1 is only 4B aligned)
    long base = (long)blockIdx.x * (BLK * NCOLS);
#pragma unroll
    for (int jj = 0; jj < NCOLS; ++jj) {
        long g = base + jj * BLK + tid;
        if (g < (long)NROWS * NCOLS) out[1 + g] = stage[jj * BLK + tid];
    }
}

// Resolve the scalar recurrence from the 8 reductions.
__global__ void np_final(const float* __restrict__ ws, float* __restrict__ out) {
    if (threadIdx.x == 0 && blockIdx.x == 0) {
        float S[TSTEPS], W[TSTEPS];
        for (int t = 0; t < TSTEPS; ++t) { S[t] = ws[WS_RED + t]; W[t] = ws[WS_RED + 4 + t]; }
        float sc[TSTEPS + 1]; sc[0] = 0.0f;
        for (int t = 0; t < TSTEPS; ++t) {
            const float* ao = ws + WS_AOP + t * NOPS;
            float s1 = sc[t];
            float s3 = (t >= 2) ? sc[t - 2] : sc[0];
            sc[t + 1] = ao[OP_SUM] * W[t] + ao[OP_COUNT] * S[t] + ao[OP_DIFF] * (s3 - s1);
        }
        out[0] = sc[TSTEPS];
    }
}

extern "C" void kernel_launch(void* const* d_in, const int* in_sizes, int n_in,
                              void* d_out, int out_size, void* d_ws, size_t ws_size,
                              hipStream_t stream) {
    const int*   iq      = (const int*)d_in[0];
    const int*   nums    = (const int*)d_in[1];
    const int*   lwi     = (const int*)d_in[2];
    const float* table   = (const float*)d_in[3];
    const float* emb     = (const float*)d_in[4];
    const float* Wx      = (const float*)d_in[5];
    const float* Wh      = (const float*)d_in[6];
    const float* W_op    = (const float*)d_in[7];
    const float* op_emb  = (const float*)d_in[8];
    const float* W_col   = (const float*)d_in[9];
    const float* col_emb = (const float*)d_in[10];
    const float* W_hist  = (const float*)d_in[11];
    const float* U       = (const float*)d_in[12];
    float* out = (float*)d_out;
    float* ws  = (float*)d_ws;

    np_init<<<1, 32, 0, stream>>>(ws);
    np_gemm_xw<<<64, 32, 0, stream>>>(iq, emb, Wx, ws);
    np_rnn<<<1, H, 0, stream>>>(Wh, ws);
    np_select<<<1, H, 0, stream>>>(nums, lwi, W_op, op_emb, W_col, col_emb, W_hist, U, ws);
    np_table<<<(NROWS + BLK - 1) / BLK, BLK, 0, stream>>>(table, ws, out);
    np_final<<<1, 32, 0, stream>>>(ws, out);
}